// PhysicalCircleLayer_5634997092696
// MI455X (gfx1250) — compile-verified
//
#include <hip/hip_runtime.h>
#include <math.h>

// ---------------------------------------------------------------------------
// PhysicalCircleLayer on MI455X (gfx1250)
// Memory-bound segmented-min over 128 x 512 x 512 f32 maps (134 MB @ 23.3TB/s
// => ~5.8us floor). No matmul content => no WMMA; the CDNA5 feature used is
// the async global->LDS copy engine (ASYNCcnt), 4-deep pipelined, with
// execz-skipped sparse updates and u32-bitcast float-min atomics.
// Each thread consumes only its own async-loaded LDS bytes, so per-wave
// s_wait_asynccnt is the only synchronization needed (no workgroup barriers
// in the streaming loop).
// ---------------------------------------------------------------------------

#define AS1 __attribute__((address_space(1)))
#define AS3 __attribute__((address_space(3)))

typedef __attribute__((ext_vector_type(4))) int v4i;

#define INFV 1000000000.0f
#define MUV 1e-8f
#define SAFE_T 0.05f
#define NPIX 512
#define PARTS 8
#define MAXPARTS 16
#define SLICES 8
#define TILE_ELEMS 1024 // 256 threads * 4 floats (16B) each
#define TILES 32        // per block chunk = 32768 elements
#define CHUNK (TILES * TILE_ELEMS)
#define DEPTH 4         // async pipeline depth (buffers)
#define TWO_PI_F 6.283185307179586f

#if defined(__HIP_DEVICE_COMPILE__) && __has_builtin(__builtin_amdgcn_global_load_async_to_lds_b128)
#define HAVE_ASYNC_B128 1
#else
#define HAVE_ASYNC_B128 0
#endif

#if defined(__HIP_DEVICE_COMPILE__) && __has_builtin(__builtin_amdgcn_s_wait_asynccnt)
#define HAVE_WAIT_ASYNC 1
#else
#define HAVE_WAIT_ASYNC 0
#endif

__device__ __forceinline__ void async_copy16(const float* gp, float* lp) {
#if HAVE_ASYNC_B128
  __builtin_amdgcn_global_load_async_to_lds_b128((AS1 v4i*)(void*)gp,
                                                 (AS3 v4i*)(void*)lp, 0, 0);
#else
  unsigned lds_off = (unsigned)(unsigned long)(AS3 char*)lp;
  unsigned long long ga = (unsigned long long)gp;
  asm volatile("global_load_async_to_lds_b128 %0, %1, off"
               :
               : "v"(lds_off), "v"(ga)
               : "memory");
#endif
}

template <int N>
__device__ __forceinline__ void s_wait_async() {
#if HAVE_WAIT_ASYNC
  __builtin_amdgcn_s_wait_asynccnt(N);
#else
  asm volatile("s_wait_asynccnt %0" : : "i"(N) : "memory");
#endif
}

// ---------------------------------------------------------------------------
// Kernel 0: init per-(batch,partition) u32 minima to bits(INF)
// ---------------------------------------------------------------------------
__global__ void pc_init(unsigned* __restrict__ gmin, int n) {
  int i = blockIdx.x * blockDim.x + threadIdx.x;
  if (i < n) gmin[i] = __float_as_uint(INFV);
}

// ---------------------------------------------------------------------------
// Kernel 1: streaming segmented-min reduce.
// grid = (B, SLICES), block = 256 (8 waves). Each block streams 128 KiB of
// seg_map through a 4 x 4 KiB LDS ring via async global->LDS copies.
// ---------------------------------------------------------------------------
__global__ __launch_bounds__(256) void pc_reduce(const float* __restrict__ seg,
                                                 const float* __restrict__ paras,
                                                 const float* __restrict__ traj,
                                                 const float* __restrict__ cpos,
                                                 unsigned* __restrict__ gmin) {
  __shared__ __align__(16) float buf[DEPTH][TILE_ELEMS];
  __shared__ unsigned smin[PARTS];

  const int b = blockIdx.x;
  const int s = blockIdx.y;
  const int tid = threadIdx.x;

  if (tid < PARTS) smin[tid] = __float_as_uint(INFV);
  __syncthreads(); // smin visible before end-of-kernel atomics

  // Per-batch uniform parameters (scalar loads).
  const float wx = paras[b * 4 + 0];
  const float wy = paras[b * 4 + 1];
  const float bx = paras[b * 4 + 2];
  const float by = paras[b * 4 + 3];
  const float cx = cpos[b * 2 + 0];
  const float cy = cpos[b * 2 + 1];
  // moving_vector = (traj[7]+cpos) - (traj[0]+cpos) = traj[7]-traj[0]
  const float mvx = traj[b * 16 + 14] - traj[b * 16 + 0];
  const float mvy = traj[b * 16 + 15] - traj[b * 16 + 1];
  const float r = 2.0f * sqrtf(mvx * mvx + mvy * mvy); // VISION_RADIUS * L
  const float r2 = r * r;

  // dvx = (x - bx)/wx - cx  ==  x*ax - cxx   (and same for y)
  const float ax = 1.0f / wx;
  const float ay = 1.0f / wy;
  const float cxx = bx * ax + cx;
  const float cyy = by * ay + cy;

  const size_t chunk0 = (size_t)b * (size_t)(NPIX * NPIX) + (size_t)s * CHUNK;
  const float* gbase = seg + chunk0 + (size_t)tid * 4;

  float acc[PARTS];
#pragma unroll
  for (int k = 0; k < PARTS; ++k) acc[k] = INFV;

  // Process one landed tile (thread reads back only its own 16 bytes).
  auto process = [&](int t) {
    const float4 m4 = *(const float4*)&buf[t & (DEPTH - 1)][tid * 4];
    const int e = s * CHUNK + t * TILE_ELEMS + tid * 4; // flat pixel index
    const float x = (float)(e >> 9);                    // row (constant over 4)
    const float y0 = (float)(e & 511);                  // col of element 0
    const float dvx = x * ax - cxx;
    const float dvx2 = dvx * dvx;
    const float dvy0 = y0 * ay - cyy;

    const float mj[4] = {m4.x, m4.y, m4.z, m4.w};
#pragma unroll
    for (int j = 0; j < 4; ++j) {
      const float dvy = dvy0 + (float)j * ay;
      const float d2 = fmaf(dvy, dvy, dvx2);
      const float m = mj[j];
      // ~0.5% of pixels pass: divergent block is skipped via s_cbranch_execz
      if (m > SAFE_T && d2 <= r2) {
        const float d = (sqrtf(d2) + MUV) / (m + MUV);
        // bin = octant of atan2(dvx, dvy) mapped to [0, 2pi) / (pi/4).
        const bool an = dvx >= 0.0f;
        const bool bp = dvy > 0.0f;
        const int Q = an ? (bp ? 0 : 1) : (bp ? 3 : 2);
        const float aa = fabsf(dvx);
        const float ab = fabsf(dvy);
        const bool w = (Q & 1) ? (aa <= ab) : (aa >= ab);
        const int bin = 2 * Q + (w ? 1 : 0);
#pragma unroll
        for (int k = 0; k < PARTS; ++k)
          if (bin == k) acc[k] = fminf(acc[k], d);
      }
    }
  };

  // Prime the pipeline: tiles 0..DEPTH-2
#pragma unroll
  for (int i = 0; i < DEPTH - 1; ++i)
    async_copy16(gbase + i * TILE_ELEMS, &buf[i][tid * 4]);

  int t = 0;
  for (; t < TILES - (DEPTH - 1); ++t) {
    async_copy16(gbase + (t + DEPTH - 1) * TILE_ELEMS,
                 &buf[(t + DEPTH - 1) & (DEPTH - 1)][tid * 4]);
    s_wait_async<DEPTH - 1>(); // tile t has landed for this wave
    process(t);
  }
  // Drain: DEPTH-1 tiles still in flight.
  s_wait_async<2>();
  process(t++);
  s_wait_async<1>();
  process(t++);
  s_wait_async<0>();
  process(t);

  // Wave32 reduce, then LDS u32 atomic-min (valid since all d >= 0),
  // then one global atomic-min per (block, partition).
#pragma unroll
  for (int k = 0; k < PARTS; ++k) {
    float v = acc[k];
#pragma unroll
    for (int off = 16; off > 0; off >>= 1) v = fminf(v, __shfl_xor(v, off, 32));
    if ((tid & 31) == 0) atomicMin(&smin[k], __float_as_uint(v));
  }
  __syncthreads();
  if (tid < PARTS) atomicMin(&gmin[b * PARTS + tid], smin[tid]);
}

// ---------------------------------------------------------------------------
// Kernel 2: finalize features into (B, 16, 3)
// ---------------------------------------------------------------------------
__global__ void pc_final(const unsigned* __restrict__ gmin,
                         const float* __restrict__ traj,
                         float* __restrict__ out) {
  const int b = blockIdx.x;
  const int p = threadIdx.x; // 0..15
  const float mvx = traj[b * 16 + 14] - traj[b * 16 + 0];
  const float mvy = traj[b * 16 + 15] - traj[b * 16 + 1];
  const float L = sqrtf(mvx * mvx + mvy * mvy);
  float v0 = 0.0f, v1 = 0.0f, v2 = 0.0f;
  if (p < PARTS) {
    const float md = __uint_as_float(gmin[b * PARTS + p]);
    const float obs = (md < INFV) ? 1.0f : 0.0f;
    v0 = L * obs;
    v1 = md * obs;
    v2 = (TWO_PI_F * ((float)p + 0.5f) / (float)PARTS) * obs;
  }
  out[b * (MAXPARTS * 3) + p * 3 + 0] = v0;
  out[b * (MAXPARTS * 3) + p * 3 + 1] = v1;
  out[b * (MAXPARTS * 3) + p * 3 + 2] = v2;
}

extern "C" void kernel_launch(void* const* d_in, const int* in_sizes, int n_in,
                              void* d_out, int out_size, void* d_ws, size_t ws_size,
                              hipStream_t stream) {
  const float* seg = (const float*)d_in[0];   // (B, 512, 512)
  const float* paras = (const float*)d_in[1]; // (B, 4)
  const float* traj = (const float*)d_in[2];  // (B, 8, 2)
  const float* cpos = (const float*)d_in[3];  // (B, 1, 2)
  // d_in[4] = map_pos_pixel: regenerated analytically, not read.
  float* out = (float*)d_out;

  const int B = in_sizes[1] / 4;
  unsigned* gmin = (unsigned*)d_ws; // B * PARTS u32 minima

  pc_init<<<(B * PARTS + 255) / 256, 256, 0, stream>>>(gmin, B * PARTS);
  dim3 grid(B, SLICES);
  pc_reduce<<<grid, 256, 0, stream>>>(seg, paras, traj, cpos, gmin);
  pc_final<<<B, MAXPARTS, 0, stream>>>(gmin, traj, out);
}